// InteractionLayer_54202487275607
// MI455X (gfx1250) — compile-verified
//
#include <hip/hip_runtime.h>
#include <hip/hip_bf16.h>
#include <math.h>

#define HEADS 4
#define DIM 128
#define NEG_SLOPE 0.2f
#define LN_EPS 1e-5f
#define RPB 15              // row tiles per GEMM block

typedef float v2f __attribute__((ext_vector_type(2)));
typedef float v8f __attribute__((ext_vector_type(8)));

static __device__ __forceinline__ float lrelu(float x) {
    return x > 0.0f ? x : NEG_SLOPE * x;
}
// order-preserving float <-> uint mapping for atomicMax on floats
static __device__ __forceinline__ unsigned ford(float f) {
    unsigned u = __float_as_uint(f);
    return (u & 0x80000000u) ? ~u : (u | 0x80000000u);
}
static __device__ __forceinline__ float funord(unsigned v) {
    unsigned u = (v & 0x80000000u) ? (v ^ 0x80000000u) : ~v;
    return __uint_as_float(u);
}
static __device__ __forceinline__ float wred32(float v) {
#pragma unroll
    for (int off = 16; off > 0; off >>= 1) v += __shfl_xor(v, off, 32);
    return v;
}

// ---------------------------------------------------------------------------
// K1: 5 GEMMs [N x 128] x [128 x 128] via V_WMMA_F32_16X16X4_F32 (wave32).
// B-register-resident: each wave keeps its 128x16 W column tile in 32x v2f
// (64 VGPRs), then sweeps RPB row tiles of x double-buffered through LDS.
// blockIdx.x = group of RPB row tiles, blockIdx.y = matrix (0..3 GAT, 4 GCN).
// ---------------------------------------------------------------------------
__global__ __launch_bounds__(256) void k_gemm_wmma(
    const float* __restrict__ x, const float* __restrict__ W_gat,
    const float* __restrict__ W_gcn, float* __restrict__ xh,
    float* __restrict__ xg, int N, int ntiles)
{
    __shared__ float As[2][16][DIM + 4];   // pad 132: conflict-free column reads
    const int tid = threadIdx.x;
    const int mat = blockIdx.y;
    const int t0  = blockIdx.x * RPB;
    if (t0 >= ntiles) return;
    const int t1  = (t0 + RPB < ntiles) ? (t0 + RPB) : ntiles;

    const float* __restrict__ W   = (mat < HEADS) ? (W_gat + (size_t)mat * DIM * DIM) : W_gcn;
    float*       __restrict__ dst = (mat < HEADS) ? (xh + (size_t)mat * N * DIM) : xg;

    const int wave  = tid >> 5;
    const int lane  = tid & 31;
    const int n0    = wave * 16;          // column tile of this wave
    const int mn    = lane & 15;          // A row / B,C col within tile
    const int koff  = (lane >> 4) * 2;    // K pair: lanes 0-15 -> {0,1}, 16-31 -> {2,3}
    const int rbase = (lane >> 4) * 8;    // C rows r / r+8 split at lane 16

    // ---- B column tile resident in registers: 32 x v2f = 64 VGPRs ----
    v2f bfrag[DIM / 4];
#pragma unroll
    for (int i = 0; i < DIM / 4; ++i) {
        int k = i * 4 + koff;
        bfrag[i].x = W[(size_t)k * DIM + n0 + mn];
        bfrag[i].y = W[(size_t)(k + 1) * DIM + n0 + mn];
    }

    // ---- stage first A tile ----
    {
        int row0 = t0 * 16;
        for (int i = tid; i < 16 * DIM / 4; i += 256) {
            int r = i >> 5, c4 = (i & 31) * 4;
            int gr = row0 + r;
            float4 v = make_float4(0.f, 0.f, 0.f, 0.f);
            if (gr < N) v = *(const float4*)&x[(size_t)gr * DIM + c4];
            *(float4*)&As[0][r][c4] = v;
        }
    }
    __syncthreads();

    for (int rt = t0; rt < t1; ++rt) {
        const int buf = (rt - t0) & 1;
        // prefetch next A tile into the other buffer (overlaps with WMMA)
        if (rt + 1 < t1) {
            int row0 = (rt + 1) * 16;
            for (int i = tid; i < 16 * DIM / 4; i += 256) {
                int r = i >> 5, c4 = (i & 31) * 4;
                int gr = row0 + r;
                float4 v = make_float4(0.f, 0.f, 0.f, 0.f);
                if (gr < N) v = *(const float4*)&x[(size_t)gr * DIM + c4];
                *(float4*)&As[buf ^ 1][r][c4] = v;
            }
        }

        v8f c = {};
#pragma unroll
        for (int i = 0; i < DIM / 4; ++i) {
            v2f a = *(const v2f*)&As[buf][mn][i * 4 + koff];
            c = __builtin_amdgcn_wmma_f32_16x16x4_f32(false, a, false, bfrag[i],
                                                      (short)0, c, false, false);
        }

        int row0 = rt * 16;
#pragma unroll
        for (int r = 0; r < 8; ++r) {
            int gr = row0 + rbase + r;
            if (gr < N) dst[(size_t)gr * DIM + n0 + mn] = c[r];
        }
        __syncthreads();   // next iteration's compute buffer fully prefetched
    }
}

// ---------------------------------------------------------------------------
// K2: a_s[n,h] = xh[h,n,:].att_src[h];  a_d likewise. One thread per (n,h).
// ---------------------------------------------------------------------------
__global__ __launch_bounds__(256) void k_attn_dots(
    const float* __restrict__ xh, const float* __restrict__ att_src,
    const float* __restrict__ att_dst, float* __restrict__ a_s,
    float* __restrict__ a_d, int N)
{
    int idx = blockIdx.x * 256 + threadIdx.x;
    if (idx >= N * HEADS) return;
    int h = idx & (HEADS - 1);
    int n = idx >> 2;
    const float4* xp = (const float4*)(xh + ((size_t)h * N + n) * DIM);
    const float4* sp = (const float4*)(att_src + h * DIM);
    const float4* dp = (const float4*)(att_dst + h * DIM);
    float s = 0.0f, d = 0.0f;
#pragma unroll 8
    for (int i = 0; i < DIM / 4; ++i) {
        float4 v = xp[i], a = sp[i], b = dp[i];
        s += v.x * a.x + v.y * a.y + v.z * a.z + v.w * a.w;
        d += v.x * b.x + v.y * b.y + v.z * b.z + v.w * b.w;
    }
    a_s[(size_t)n * HEADS + h] = s;
    a_d[(size_t)n * HEADS + h] = d;
}

// K3: seed segment-max with the self-loop edge, deg = 1 (self loop).
__global__ __launch_bounds__(256) void k_node_init(
    const float* __restrict__ a_s, const float* __restrict__ a_d,
    unsigned* __restrict__ m_u, unsigned* __restrict__ deg, int N)
{
    int idx = blockIdx.x * 256 + threadIdx.x;
    if (idx >= N * HEADS) return;
    float e = lrelu(a_s[idx] + a_d[idx]);
    m_u[idx] = ford(e);
    if ((idx & (HEADS - 1)) == 0) deg[idx >> 2] = 1u;
}

__global__ __launch_bounds__(256) void k_zero_acc(
    float* __restrict__ gat_acc, float* __restrict__ gcn_acc, int n)
{
    int i = blockIdx.x * 256 + threadIdx.x;
    if (i < n) { gat_acc[i] = 0.0f; gcn_acc[i] = 0.0f; }
}

// K4: per real edge: segment max (ordered-uint atomicMax) + in-degree count.
__global__ __launch_bounds__(256) void k_edge_max(
    const int* __restrict__ ei, const float* __restrict__ a_s,
    const float* __restrict__ a_d, unsigned* __restrict__ m_u,
    unsigned* __restrict__ deg, int E)
{
    int e = blockIdx.x * 256 + threadIdx.x;
    if (e >= E) return;
    int s = ei[e], d = ei[E + e];
#pragma unroll
    for (int h = 0; h < HEADS; ++h) {
        float v = lrelu(a_s[s * HEADS + h] + a_d[d * HEADS + h]);
        atomicMax(&m_u[d * HEADS + h], ford(v));
    }
    atomicAdd(&deg[d], 1u);
}

// K5: denom starts with self-loop contribution; dinv = rsqrt(deg).
__global__ __launch_bounds__(256) void k_node_mid(
    const float* __restrict__ a_s, const float* __restrict__ a_d,
    const unsigned* __restrict__ m_u, const unsigned* __restrict__ deg,
    float* __restrict__ denom, float* __restrict__ dinv, int N)
{
    int idx = blockIdx.x * 256 + threadIdx.x;
    if (idx >= N * HEADS) return;
    float e = lrelu(a_s[idx] + a_d[idx]);
    denom[idx] = expf(e - funord(m_u[idx]));
    if ((idx & (HEADS - 1)) == 0) dinv[idx >> 2] = rsqrtf((float)deg[idx >> 2]);
}

// K6: accumulate softmax denominators over real edges.
__global__ __launch_bounds__(256) void k_edge_denom(
    const int* __restrict__ ei, const float* __restrict__ a_s,
    const float* __restrict__ a_d, const unsigned* __restrict__ m_u,
    float* __restrict__ denom, int E)
{
    int e = blockIdx.x * 256 + threadIdx.x;
    if (e >= E) return;
    int s = ei[e], d = ei[E + e];
#pragma unroll
    for (int h = 0; h < HEADS; ++h) {
        float v = lrelu(a_s[s * HEADS + h] + a_d[d * HEADS + h]);
        atomicAdd(&denom[d * HEADS + h], expf(v - funord(m_u[d * HEADS + h])));
    }
}

// K7: one wave per edge (incl. self loops). Heads are reduced in-register so
// each feature needs a single f32 atomic for GAT plus one for GCN.
__global__ __launch_bounds__(256) void k_edge_aggr(
    const int* __restrict__ ei, const float* __restrict__ a_s,
    const float* __restrict__ a_d, const unsigned* __restrict__ m_u,
    const float* __restrict__ denom, const float* __restrict__ dinv,
    const float* __restrict__ xh, const float* __restrict__ xg,
    float* __restrict__ gat_acc, float* __restrict__ gcn_acc,
    int N, int E, int ET)
{
    int gw = (blockIdx.x * 256 + threadIdx.x) >> 5;
    int lane = threadIdx.x & 31;
    if (gw >= ET) return;
    int s, d;
    if (gw < E) { s = ei[gw]; d = ei[E + gw]; } else { s = d = gw - E; }

    float coeff[HEADS];
#pragma unroll
    for (int h = 0; h < HEADS; ++h) {
        float v = lrelu(a_s[s * HEADS + h] + a_d[d * HEADS + h]);
        coeff[h] = expf(v - funord(m_u[d * HEADS + h])) / denom[d * HEADS + h] * (1.0f / HEADS);
    }
    float gco = dinv[s] * dinv[d];

#pragma unroll
    for (int ch = 0; ch < DIM / 32; ++ch) {
        int f = ch * 32 + lane;
        float acc = 0.0f;
#pragma unroll
        for (int h = 0; h < HEADS; ++h)
            acc += coeff[h] * xh[((size_t)h * N + s) * DIM + f];
        atomicAdd(&gat_acc[(size_t)d * DIM + f], acc);
        atomicAdd(&gcn_acc[(size_t)d * DIM + f], gco * xg[(size_t)s * DIM + f]);
    }
}

// K8: one wave per node: gate GEMV (256->2), softmax gate, fuse + residual,
// LayerNorm. 4 features per lane, __shfl_xor tree reductions (wave32).
__global__ __launch_bounds__(256) void k_finalize(
    const float* __restrict__ gat_acc, const float* __restrict__ gcn_acc,
    const float* __restrict__ b_gat, const float* __restrict__ b_gcn,
    const float* __restrict__ W_gate, const float* __restrict__ b_gate,
    const float* __restrict__ gamma, const float* __restrict__ beta,
    float* __restrict__ out, int N)
{
    int gw = (blockIdx.x * 256 + threadIdx.x) >> 5;
    int lane = threadIdx.x & 31;
    if (gw >= N) return;
    int f0 = lane * 4;

    float4 ga4 = *(const float4*)&gat_acc[(size_t)gw * DIM + f0];
    float4 gb4 = *(const float4*)&b_gat[f0];
    float4 gc4 = *(const float4*)&gcn_acc[(size_t)gw * DIM + f0];
    float4 cb4 = *(const float4*)&b_gcn[f0];
    float ga[4] = {ga4.x + gb4.x, ga4.y + gb4.y, ga4.z + gb4.z, ga4.w + gb4.w};
    float gc[4] = {gc4.x + cb4.x, gc4.y + cb4.y, gc4.z + cb4.z, gc4.w + cb4.w};

    float l0 = 0.0f, l1 = 0.0f;
#pragma unroll
    for (int j = 0; j < 4; ++j) {
        int dd = f0 + j;
        l0 += ga[j] * W_gate[dd * 2 + 0] + gc[j] * W_gate[(DIM + dd) * 2 + 0];
        l1 += ga[j] * W_gate[dd * 2 + 1] + gc[j] * W_gate[(DIM + dd) * 2 + 1];
    }
    l0 = wred32(l0) + b_gate[0];
    l1 = wred32(l1) + b_gate[1];
    float mx = fmaxf(l0, l1);
    float e0 = expf(l0 - mx), e1 = expf(l1 - mx);
    float inv = 1.0f / (e0 + e1);
    float g0 = e0 * inv, g1 = e1 * inv;

    float hv[4], sum = 0.0f, sq = 0.0f;
#pragma unroll
    for (int j = 0; j < 4; ++j) {
        hv[j] = g0 * ga[j] + g1 * gc[j] + ga[j];   // fused + residual(gat_out)
        sum += hv[j];
        sq  += hv[j] * hv[j];
    }
    sum = wred32(sum);
    sq  = wred32(sq);
    float mu   = sum * (1.0f / DIM);
    float var  = sq * (1.0f / DIM) - mu * mu;
    float rstd = rsqrtf(var + LN_EPS);

    float4 gm = *(const float4*)&gamma[f0];
    float4 bt = *(const float4*)&beta[f0];
    float4 o;
    o.x = (hv[0] - mu) * rstd * gm.x + bt.x;
    o.y = (hv[1] - mu) * rstd * gm.y + bt.y;
    o.z = (hv[2] - mu) * rstd * gm.z + bt.z;
    o.w = (hv[3] - mu) * rstd * gm.w + bt.w;
    *(float4*)&out[(size_t)gw * DIM + f0] = o;
}

extern "C" void kernel_launch(void* const* d_in, const int* in_sizes, int n_in,
                              void* d_out, int out_size, void* d_ws, size_t ws_size,
                              hipStream_t stream)
{
    const float* x       = (const float*)d_in[0];
    const int*   ei      = (const int*)d_in[1];     // [2][E], row0=src, row1=dst
    const float* W_gat   = (const float*)d_in[2];
    const float* att_src = (const float*)d_in[3];
    const float* att_dst = (const float*)d_in[4];
    const float* b_gat   = (const float*)d_in[5];
    const float* W_gcn   = (const float*)d_in[6];
    const float* b_gcn   = (const float*)d_in[7];
    const float* W_gate  = (const float*)d_in[8];
    const float* b_gate  = (const float*)d_in[9];
    const float* gamma   = (const float*)d_in[10];
    const float* beta    = (const float*)d_in[11];

    const int N  = in_sizes[0] / DIM;
    const int E  = in_sizes[1] / 2;
    const int ET = E + N;                            // real edges + self loops

    char* ws = (char*)d_ws;
    size_t off = 0;
    auto carve = [&](size_t bytes) -> void* {
        void* p = ws + off;
        off += (bytes + 255) & ~(size_t)255;
        return p;
    };
    float*    xh      = (float*)   carve((size_t)HEADS * N * DIM * sizeof(float)); // [H][N][D]
    float*    xg      = (float*)   carve((size_t)N * DIM * sizeof(float));
    float*    a_s     = (float*)   carve((size_t)N * HEADS * sizeof(float));
    float*    a_d     = (float*)   carve((size_t)N * HEADS * sizeof(float));
    unsigned* m_u     = (unsigned*)carve((size_t)N * HEADS * sizeof(unsigned));
    float*    denom   = (float*)   carve((size_t)N * HEADS * sizeof(float));
    unsigned* deg     = (unsigned*)carve((size_t)N * sizeof(unsigned));
    float*    dinv    = (float*)   carve((size_t)N * sizeof(float));
    float*    gat_acc = (float*)   carve((size_t)N * DIM * sizeof(float));
    float*    gcn_acc = (float*)   carve((size_t)N * DIM * sizeof(float));
    (void)ws_size; (void)n_in; (void)out_size;

    const int ntiles = (N + 15) / 16;
    dim3 gemm_grid((ntiles + RPB - 1) / RPB, HEADS + 1);
    k_gemm_wmma<<<gemm_grid, 256, 0, stream>>>(x, W_gat, W_gcn, xh, xg, N, ntiles);
    k_attn_dots<<<(N * HEADS + 255) / 256, 256, 0, stream>>>(xh, att_src, att_dst, a_s, a_d, N);
    k_node_init<<<(N * HEADS + 255) / 256, 256, 0, stream>>>(a_s, a_d, m_u, deg, N);
    k_zero_acc<<<(N * DIM + 255) / 256, 256, 0, stream>>>(gat_acc, gcn_acc, N * DIM);
    k_edge_max<<<(E + 255) / 256, 256, 0, stream>>>(ei, a_s, a_d, m_u, deg, E);
    k_node_mid<<<(N * HEADS + 255) / 256, 256, 0, stream>>>(a_s, a_d, m_u, deg, denom, dinv, N);
    k_edge_denom<<<(E + 255) / 256, 256, 0, stream>>>(ei, a_s, a_d, m_u, denom, E);
    k_edge_aggr<<<((unsigned)ET * 32 + 255) / 256, 256, 0, stream>>>(
        ei, a_s, a_d, m_u, denom, dinv, xh, xg, gat_acc, gcn_acc, N, E, ET);
    k_finalize<<<((unsigned)N * 32 + 255) / 256, 256, 0, stream>>>(
        gat_acc, gcn_acc, b_gat, b_gcn, W_gate, b_gate, gamma, beta, (float*)d_out, N);
}